// TripletLoss_4157528343144
// MI455X (gfx1250) — compile-verified
//
#include <hip/hip_runtime.h>
#include <hip/hip_bf16.h>

// ---------------------------------------------------------------------------
// TripletLoss on MI455X (gfx1250, wave32)
//   loss = mean over positive pairs (i,j) of relu(0.5 + D2[i,j] - D2[i,neg(i,j)])
//   D2[i,j] = ||x_i||^2 + ||x_j||^2 - 2 * <x_i, x_j>
// Gram matrix via V_WMMA_F32_16X16X4_F32 (native f32 matrix path, no downcast).
// LDS staging via GLOBAL_LOAD_ASYNC_TO_LDS_B128 (ASYNCcnt-tracked DMA).
// Epilogue: accumulators parked in (dead) B-panel LDS -> compact single-copy loop.
// ---------------------------------------------------------------------------

#define NROWS 8192
#define NDIM  128          // 32 float4 per row
#define TILE  128          // tile side per workgroup
#define LDSTR 132          // padded LDS row stride (floats): 132%4==0 (16B-aligned
                           // float4/async-b128 stores), (r*132)%64 spreads banks
                           // 4 apart -> conflict-free ds_load_b64 fragment reads

typedef float v2f __attribute__((ext_vector_type(2)));
typedef float v8f __attribute__((ext_vector_type(8)));

// CDNA5 async global->LDS copy (GLOBAL_LOAD_ASYNC_TO_LDS_B128, ASYNCcnt-tracked)
#if defined(__has_builtin)
#if __has_builtin(__builtin_amdgcn_global_load_async_to_lds_b128)
#define HAVE_ASYNC_COPY 1
#endif
#endif

// Exact builtin parameter types (from hipcc diagnostic):
//   param0: __attribute__((__vector_size__(16))) int __device__ *   (AS1 global)
//   param1: same vector type in AS3 (LDS)
typedef __attribute__((__vector_size__(16))) int i32x4;
typedef __attribute__((address_space(1))) i32x4* g_b128_t;
typedef __attribute__((address_space(3))) i32x4* l_b128_t;

__device__ __forceinline__ void wait_asynccnt0() {
#if defined(__has_builtin) && __has_builtin(__builtin_amdgcn_s_wait_asynccnt)
    __builtin_amdgcn_s_wait_asynccnt(0);
#else
    asm volatile("s_wait_asynccnt 0x0" ::: "memory");
#endif
}

// ---------------- kernel 1: per-row squared norms --------------------------
__global__ __launch_bounds__(256) void sq_kernel(const float* __restrict__ X,
                                                 float* __restrict__ sq) {
    int i = blockIdx.x * blockDim.x + threadIdx.x;   // 32 blocks * 256 = 8192
    const float4* x4 = (const float4*)(X + (size_t)i * NDIM);
    float s = 0.f;
#pragma unroll 8
    for (int d = 0; d < NDIM / 4; ++d) {
        float4 v = x4[d];
        s = fmaf(v.x, v.x, s);
        s = fmaf(v.y, v.y, s);
        s = fmaf(v.z, v.z, s);
        s = fmaf(v.w, v.w, s);
    }
    sq[i] = s;
}

// counter-based hash (PCG output permutation) -> deterministic per (i,j)
__device__ __forceinline__ unsigned pcg_step(unsigned& s) {
    s = s * 747796405u + 2891336453u;
    unsigned w = ((s >> ((s >> 28u) + 4u)) ^ s) * 277803737u;
    return (w >> 22u) ^ w;
}

// ---------------- kernel 2: fused Gram-tile + hinge loss -------------------
__global__ __launch_bounds__(256) void tile_kernel(const float* __restrict__ X,
                                                   const int*   __restrict__ lab,
                                                   const float* __restrict__ sq,
                                                   float* __restrict__ pH,
                                                   float* __restrict__ pW) {
    __shared__ float lA[TILE][LDSTR];   // 67.6 KB (row panel; x_i rows for dots too)
    __shared__ float lB[TILE][LDSTR];   // 67.6 KB (col panel; reused as Gram tile)
    __shared__ int   labR[TILE], labC[TILE];
    __shared__ float sqR[TILE], sqC[TILE];
    __shared__ float redH[8], redW[8];

    const int tid  = threadIdx.x;
    const int bm   = blockIdx.y, bn = blockIdx.x;
    const int row0 = bm * TILE, col0 = bn * TILE;

    // ---- stage both 128x128 f32 panels into LDS + tile labels/norms --------
    const float4* X4 = (const float4*)X;
#if HAVE_ASYNC_COPY
    // async DMA path: no VGPR round-trip, tracked by ASYNCcnt
    for (int idx = tid; idx < TILE * (NDIM / 4); idx += 256) {
        int r  = idx >> 5;
        int c4 = idx & 31;
        __builtin_amdgcn_global_load_async_to_lds_b128(
            (g_b128_t)(X4 + (size_t)(row0 + r) * (NDIM / 4) + c4),
            (l_b128_t)&lA[r][c4 * 4], 0, 0);
        __builtin_amdgcn_global_load_async_to_lds_b128(
            (g_b128_t)(X4 + (size_t)(col0 + r) * (NDIM / 4) + c4),
            (l_b128_t)&lB[r][c4 * 4], 0, 0);
    }
#else
    for (int idx = tid; idx < TILE * (NDIM / 4); idx += 256) {
        int r  = idx >> 5;
        int c4 = idx & 31;
        float4 va = X4[(size_t)(row0 + r) * (NDIM / 4) + c4];
        float4 vb = X4[(size_t)(col0 + r) * (NDIM / 4) + c4];
        *(float4*)&lA[r][c4 * 4] = va;
        *(float4*)&lB[r][c4 * 4] = vb;
    }
#endif
    if (tid < TILE) {
        labR[tid] = lab[row0 + tid];
        labC[tid] = lab[col0 + tid];
        sqR[tid]  = sq[row0 + tid];
        sqC[tid]  = sq[col0 + tid];
    }
#if HAVE_ASYNC_COPY
    wait_asynccnt0();
#endif
    __syncthreads();

    const int wave = tid >> 5;          // 8 waves, each owns 16 rows x 128 cols
    const int lane = tid & 31;
    const int half = lane >> 4;         // K-half select per ISA 16x4 A layout
    const int lr   = lane & 15;         // M (A) / N (B) index within 16
    const int rl0  = wave * 16;         // first local row of this wave's strip

    v8f acc[8];
    const v8f vzero = {0.f, 0.f, 0.f, 0.f, 0.f, 0.f, 0.f, 0.f};
#pragma unroll
    for (int ct = 0; ct < 8; ++ct) acc[ct] = vzero;

    // ---- K loop: 32 steps of V_WMMA_F32_16X16X4_F32 per column tile --------
    // A frag (16x4, MxK): lanes 0-15 -> M=lr, K=kk+{0,1}; lanes 16-31 -> K=kk+{2,3}
    // B frag (4x16, KxN) for X*X^T loads with the identical pattern on the col panel.
    for (int kk = 0; kk < NDIM; kk += 4) {
        v2f a = *(const v2f*)&lA[rl0 + lr][kk + 2 * half];
#pragma unroll
        for (int ct = 0; ct < 8; ++ct) {
            v2f b = *(const v2f*)&lB[ct * 16 + lr][kk + 2 * half];
            acc[ct] = __builtin_amdgcn_wmma_f32_16x16x4_f32(
                false, a, false, b, (short)0, acc[ct], false, false);
        }
    }

    // ---- park accumulators in LDS (lB panel is dead after the K loop) ------
    // C/D layout: acc[ct][v] = G[rl0 + v + 8*half][ct*16 + lr]; v=0..7 are
    // consecutive rows of one column -> column-major Dt makes each acc[ct] a
    // single contiguous 32B LDS store.
    __syncthreads();                    // all waves done reading lB
    float (*Dt)[LDSTR] = lB;            // Dt[col_local][row_local]
#pragma unroll
    for (int ct = 0; ct < 8; ++ct) {
        *(v8f*)&Dt[ct * 16 + lr][rl0 + 8 * half] = acc[ct];
    }
    __syncthreads();

    // ---- compact epilogue: 64 elements per thread, single code copy --------
    // Thread owns fixed column jl = tid&127; rows il = 2e + (tid>>7).
    float sh = 0.f, sw = 0.f;
    const int jl  = tid & 127;
    const int j   = col0 + jl;
    const int ljv = labC[jl];
    const float sqj = sqC[jl];
    for (int e = 0; e < 64; ++e) {
        const int il = 2 * e + (tid >> 7);
        const int i  = row0 + il;
        if (i == j || labR[il] != ljv) continue;   // not a positive pair

        const float g    = Dt[jl][il];
        const float sqi  = sqR[il];
        const float dpos = sqi + sqj - 2.f * g;

        // deterministic negative: rejection-sample k with lab[k] != ljv
        unsigned s = ((unsigned)i * 8192u + (unsigned)j) ^ 0x9E3779B9u;
        int k = (i == 0) ? 1 : 0;            // (unreachable-in-practice fallback)
        for (int t = 0; t < 64; ++t) {
            int cand = (int)(pcg_step(s) & (NROWS - 1));
            if (lab[cand] != ljv) { k = cand; break; }
        }

        // d_neg = ||x_i||^2 + ||x_k||^2 - 2<x_i,x_k>; x_i from LDS (wave-uniform
        // row -> broadcast reads), x_k gathered from L2-resident X
        const float4* xk = (const float4*)(X + (size_t)k * NDIM);
        float dot = 0.f;
#pragma unroll 8
        for (int d4 = 0; d4 < NDIM / 4; ++d4) {
            float4 xi = *(const float4*)&lA[il][d4 * 4];
            float4 xv = xk[d4];
            dot = fmaf(xi.x, xv.x, dot);
            dot = fmaf(xi.y, xv.y, dot);
            dot = fmaf(xi.z, xv.z, dot);
            dot = fmaf(xi.w, xv.w, dot);
        }
        const float dneg = sqi + sq[k] - 2.f * dot;
        const float h    = 0.5f + dpos - dneg;
        sh += (h > 0.f) ? h : 0.f;
        sw += 1.f;
    }

    // ---- wave32 reduction, then per-tile partial ---------------------------
#pragma unroll
    for (int off = 16; off > 0; off >>= 1) {
        sh += __shfl_xor(sh, off, 32);
        sw += __shfl_xor(sw, off, 32);
    }
    if (lane == 0) { redH[wave] = sh; redW[wave] = sw; }
    __syncthreads();
    if (tid == 0) {
        float H = 0.f, W = 0.f;
#pragma unroll
        for (int w2 = 0; w2 < 8; ++w2) { H += redH[w2]; W += redW[w2]; }
        const int tileId = blockIdx.y * gridDim.x + blockIdx.x;
        pH[tileId] = H;
        pW[tileId] = W;
    }
}

// ---------------- kernel 3: deterministic final reduction ------------------
__global__ __launch_bounds__(256) void reduce_kernel(const float* __restrict__ pH,
                                                     const float* __restrict__ pW,
                                                     float* __restrict__ out,
                                                     int ntiles) {
    __shared__ float sH[256], sW[256];
    const int t = threadIdx.x;
    float h = 0.f, w = 0.f;
    for (int idx = t; idx < ntiles; idx += 256) { h += pH[idx]; w += pW[idx]; }
    sH[t] = h; sW[t] = w;
    __syncthreads();
    for (int off = 128; off > 0; off >>= 1) {
        if (t < off) { sH[t] += sH[t + off]; sW[t] += sW[t + off]; }
        __syncthreads();
    }
    if (t == 0) out[0] = sH[0] / sW[0];
}

// ---------------------------------------------------------------------------
extern "C" void kernel_launch(void* const* d_in, const int* in_sizes, int n_in,
                              void* d_out, int out_size, void* d_ws, size_t ws_size,
                              hipStream_t stream) {
    const float* X   = (const float*)d_in[0];   // Xemb [8192,128] f32
    // d_in[1] = scores (unused by the loss)
    const int*   lab = (const int*)d_in[2];     // labels [8192]
    float*       out = (float*)d_out;

    float* ws = (float*)d_ws;
    float* sq = ws;                 // 8192 floats
    float* pH = ws + NROWS;         // 4096 floats (tile partial hinge sums)
    float* pW = ws + NROWS + 4096;  // 4096 floats (tile partial pair counts)

    sq_kernel<<<NROWS / 256, 256, 0, stream>>>(X, sq);

    dim3 grid(NROWS / TILE, NROWS / TILE);      // 64 x 64 = 4096 tiles
    tile_kernel<<<grid, 256, 0, stream>>>(X, lab, sq, pH, pW);

    reduce_kernel<<<1, 256, 0, stream>>>(pH, pW, out, 4096);
}